// NearestNeighbor_24910810317191
// MI455X (gfx1250) — compile-verified
//
#include <hip/hip_runtime.h>

// -----------------------------------------------------------------------------
// 1-NN pose predictor for MI455X (gfx1250, wave32).
//
// Problem sizes (from reference): N=100000 rows, D=990=(16-1)*66, T_PRED=25, DIM=66.
// Bandwidth-bound: 396 MB of train_vel @ 23.3 TB/s ~= 17 us. We keep fp32 end to
// end (argmin is sensitive; f16/fp8 saves no HBM traffic since input is fp32).
//
// ||t-q||^2 = ||t||^2 - 2 t.q + ||q||^2 ; the constant ||q||^2 is dropped
// (argmin-invariant) and the possibly-negative score is handled by a
// sign-corrected ordered-uint key. t.q runs on the fp32 matrix pipe:
// V_WMMA_F32_16X16X4_F32 with q broadcast across all 16 B columns, so one wave
// produces 16 row-dot-products per K-sweep. ||t||^2 rides along on VALU FMAs
// (two independent accumulator chains) using the same registers -> zero extra
// memory traffic. train_vel is streamed with TH=NT (one-pass, 2x the L2 size).
// -----------------------------------------------------------------------------

typedef float v8f __attribute__((ext_vector_type(8)));
typedef float v2f __attribute__((ext_vector_type(2)));

#define N_ROWS 100000
#define DVEC   990          // (T_OBS-1)*DIM
#define T_PRED 25
#define DIM    66
#define ROWS_PER_WAVE  16
#define WAVES_PER_BLK  8
#define ROWS_PER_BLK   (ROWS_PER_WAVE * WAVES_PER_BLK)   // 128

// --- pass 0: reset the (key,index) slot every launch (ws is poisoned once, and
// graph replays must be self-initializing) --------------------------------------
__global__ void nn_init_kernel(unsigned long long* __restrict__ best) {
    if (threadIdx.x == 0) best[0] = ~0ull;
}

// --- pass 1: fused distance + argmin --------------------------------------------
__global__ __launch_bounds__(256)
void nn_dist_kernel(const float* __restrict__ obs,     // (1,16,66) flat
                    const float* __restrict__ train,   // (N, 990)
                    unsigned long long* __restrict__ best) {
    __shared__ float qs[DVEC + 2];                     // q = in_vel, zero-padded

    const int tid = threadIdx.x;
    for (int j = tid; j < DVEC; j += 256) qs[j] = obs[j + DIM] - obs[j];
    if (tid < 2) qs[DVEC + tid] = 0.0f;
    __syncthreads();

    const int wave = tid >> 5;
    const int lane = tid & 31;
    const int h    = lane >> 4;        // K-half (ISA A layout: lanes 16-31 hold K=2,3)
    const int mrow = lane & 15;        // row of the 16-row tile this lane streams

    const long long base =
        (long long)(blockIdx.x * WAVES_PER_BLK + wave) * ROWS_PER_WAVE;
    if (base >= N_ROWS) return;        // wave-uniform: EXEC stays all-ones for WMMA

    const float* rowp = train + (base + mrow) * (long long)DVEC;

    v8f  c = {};                       // D[m][n] accumulators, all n columns == t_m . q
    float nrm0 = 0.0f, nrm1 = 0.0f;    // two independent ||t_row||^2 partial chains

    int k0 = 0;
    for (; k0 + 4 <= DVEC; k0 += 4) {  // 247 full K-slices
        // streaming, one-pass: non-temporal global_load_b64 (don't thrash L2)
        v2f a = __builtin_nontemporal_load((const v2f*)(rowp + k0 + 2 * h));
        v2f b;
        b.x = qs[k0 + 2 * h];                          // B: q[k] broadcast over N
        b.y = qs[k0 + 2 * h + 1];
        nrm0 = fmaf(a.x, a.x, nrm0);
        nrm1 = fmaf(a.y, a.y, nrm1);
        c = __builtin_amdgcn_wmma_f32_16x16x4_f32(
                false, a, false, b, (short)0, c, false, false);
    }
    {   // tail slice: k=988,989 valid (h==0), k=990,991 zero-padded (h==1)
        v2f a, b;
        if (h == 0) {
            a = __builtin_nontemporal_load((const v2f*)(rowp + k0));
            b.x = qs[k0]; b.y = qs[k0 + 1];
        } else {
            a.x = 0.0f; a.y = 0.0f; b.x = 0.0f; b.y = 0.0f;
        }
        nrm0 = fmaf(a.x, a.x, nrm0);
        nrm1 = fmaf(a.y, a.y, nrm1);
        c = __builtin_amdgcn_wmma_f32_16x16x4_f32(
                false, a, false, b, (short)0, c, false, false);
    }

    // full ||t_row||^2: partner lane (lane^16) streamed the other K-half
    const float nrm     = nrm0 + nrm1;
    const float nrm_full = nrm + __shfl_xor(nrm, 16, 32);

    // Owner lanes {0..7} U {24..31}: component (lane&7) of the accumulator holds
    // D[M][*] with M = (lane&7) + 8*(lane>=16) == this lane's streamed row,
    // independent of the B/D column mapping (B is column-broadcast).
    unsigned long long key = ~0ull;
    const bool owner = (lane < 8) || (lane >= 24);
    if (owner) {
        const int ci = lane & 7;
        float dot = (ci == 0) ? c[0] : (ci == 1) ? c[1] : (ci == 2) ? c[2] :
                    (ci == 3) ? c[3] : (ci == 4) ? c[4] : (ci == 5) ? c[5] :
                    (ci == 6) ? c[6] : c[7];
        const float dis = nrm_full - 2.0f * dot;       // + ||q||^2 (const) dropped
        const long long row = base + ((lane < 8) ? lane : lane - 16);
        if (row < N_ROWS) {
            unsigned int u = __float_as_uint(dis);     // total order incl. negatives
            u = (u & 0x80000000u) ? ~u : (u | 0x80000000u);
            key = ((unsigned long long)u << 32) | (unsigned int)row;
        }
    }
    // wave-wide min of (key, index): ties -> lowest index, matching jnp.argmin
    #pragma unroll
    for (int s = 16; s >= 1; s >>= 1) {
        unsigned long long o = __shfl_xor(key, s, 32);
        key = (o < key) ? o : key;
    }
    if (lane == 0) atomicMin(best, key);
}

// --- pass 2: gather winner, copy pred_vel, cumsum -> pred_pose -------------------
__global__ void nn_finish_kernel(const float* __restrict__ obs,
                                 const float* __restrict__ train_pred_vel, // (N,25,66)
                                 const unsigned long long* __restrict__ best,
                                 float* __restrict__ out) {   // [pred_pose | pred_vel]
    const unsigned int ind = (unsigned int)(best[0] & 0xffffffffu);
    const float* pv = train_pred_vel + (long long)ind * (T_PRED * DIM);
    const int tid = threadIdx.x;

    for (int j = tid; j < T_PRED * DIM; j += blockDim.x)
        out[T_PRED * DIM + j] = pv[j];                 // pred_vel

    if (tid < DIM) {                                   // pred_pose = last + cumsum
        float acc = obs[(16 - 1) * DIM + tid];
        #pragma unroll
        for (int t = 0; t < T_PRED; ++t) {
            acc += pv[t * DIM + tid];
            out[t * DIM + tid] = acc;
        }
    }
}

extern "C" void kernel_launch(void* const* d_in, const int* in_sizes, int n_in,
                              void* d_out, int out_size, void* d_ws, size_t ws_size,
                              hipStream_t stream) {
    (void)in_sizes; (void)n_in; (void)out_size; (void)ws_size;
    const float* obs   = (const float*)d_in[0];   // observed_pose (1,16,66)
    const float* train = (const float*)d_in[1];   // train_vel     (N,990)
    const float* tpv   = (const float*)d_in[2];   // train_pred_vel(N,25,66)
    float* out = (float*)d_out;
    unsigned long long* best = (unsigned long long*)d_ws;

    nn_init_kernel<<<1, 32, 0, stream>>>(best);
    const int nblk = (N_ROWS + ROWS_PER_BLK - 1) / ROWS_PER_BLK;   // 782
    nn_dist_kernel<<<nblk, 256, 0, stream>>>(obs, train, best);
    nn_finish_kernel<<<1, 128, 0, stream>>>(obs, tpv, best, out);
}